// SelfAttention_48112223649917
// MI455X (gfx1250) — compile-verified
//
#include <hip/hip_runtime.h>

typedef __attribute__((ext_vector_type(2))) float v2f;
typedef __attribute__((ext_vector_type(8))) float v8f;

#define NB   8192
#define SEQ  64
#define DIM  32
#define NH   4
#define DH   8
#define XS   36   // padded pitch for x tile: 144B rows -> 16B-aligned for async b128
#define XP   33   // padded pitch for Q/K/V/O tiles (bank spread)
#define AP   65   // padded pitch for [64 x 64] score tile

__device__ __forceinline__ v8f wmma4(v2f a, v2f b, v8f c) {
  // D = A(16x4,f32) * B(4x16,f32) + C(16x16,f32) — full fp32 matrix op
  return __builtin_amdgcn_wmma_f32_16x16x4_f32(
      /*neg_a=*/false, a, /*neg_b=*/false, b,
      /*c_mod=*/(short)0, c, /*reuse_a=*/false, /*reuse_b=*/false);
}

__global__ __launch_bounds__(128)
void attn_fused_kernel(const float* __restrict__ x,
                       const int*   __restrict__ attend,
                       const float* __restrict__ Wq, const float* __restrict__ bq,
                       const float* __restrict__ Wk, const float* __restrict__ bk,
                       const float* __restrict__ Wv, const float* __restrict__ bv,
                       const float* __restrict__ Wp, const float* __restrict__ bp,
                       float* __restrict__ out)
{
  __shared__ float sX[SEQ * XS];   // x tile (async-staged, 9.0 KB)
  __shared__ float sQ[SEQ * XP];
  __shared__ float sK[SEQ * XP];
  __shared__ float sV[SEQ * XP];
  __shared__ float sO[SEQ * XP];   // attn output (heads interleaved -> [S, D])
  __shared__ float sA[SEQ * AP];   // scores / attention probs (16.25 KB)

  const int b    = blockIdx.x;
  const int tid  = threadIdx.x;
  const int lane = tid & 31;
  const int wave = tid >> 5;       // 0..3 -> row-tile
  const int l16  = lane & 15;
  const int lh   = lane >> 4;      // lane half selects K sub-pair / M+8
  const int row  = wave * 16 + l16;       // A-fragment row (M = lane&15)
  const int aa   = attend[b];

  // ---- stage x[b] (64x32) into LDS via CDNA5 async copy (ASYNCcnt path) ----
  const float* xb = x + (size_t)b * (SEQ * DIM);
  {
    const unsigned ldsBase = (unsigned)(uintptr_t)(&sX[0]); // LDS offset = low 32b
    for (int c = tid; c < (SEQ * DIM) / 4; c += 128) {      // 512 x 16B chunks
      const int r   = c >> 3;                               // row (8 chunks/row)
      const int col = (c & 7) * 4;                          // starting float col
      const unsigned ldst = ldsBase + (unsigned)((r * XS + col) * sizeof(float));
      const float* gp = xb + c * 4;                         // 16B-aligned global
      asm volatile("global_load_async_to_lds_b128 %0, %1, off"
                   :: "v"(ldst), "v"(gp) : "memory");
    }
    asm volatile("s_wait_asynccnt 0x0" ::: "memory");
  }
  __syncthreads();

  // ---- QKV projections: P = X @ W^T + bias, each [64,32] ----
  const float* Wg[3] = {Wq, Wk, Wv};
  const float* bg[3] = {bq, bk, bv};
  float*       Pg[3] = {sQ, sK, sV};
#pragma unroll
  for (int p = 0; p < 3; ++p) {
    const float* W = Wg[p];
#pragma unroll
    for (int nt = 0; nt < 2; ++nt) {
      v8f c = {};
#pragma unroll
      for (int k = 0; k < 8; ++k) {            // K = 32 in steps of 4
        const int kk = k * 4 + lh * 2;
        v2f a; a.x = sX[row * XS + kk]; a.y = sX[row * XS + kk + 1];
        const int n = nt * 16 + l16;
        // B[k][n] = W[n][k]; 8B-aligned pair -> single b64 load
        const v2f bbf = *reinterpret_cast<const v2f*>(W + n * DIM + kk);
        c = wmma4(a, bbf, c);
      }
      const float bias = bg[p][nt * 16 + l16];
#pragma unroll
      for (int r = 0; r < 8; ++r)              // C: M = r + 8*lh, N = l16
        Pg[p][(wave * 16 + r + 8 * lh) * XP + nt * 16 + l16] = c[r] + bias;
    }
  }
  __syncthreads();

  const float scale = 0.35355339059327373f;    // 1/sqrt(DH)

  for (int h = 0; h < NH; ++h) {
    // ---- scores: S = Qh @ Kh^T, wave computes rows [16w,16w+16) x 64 ----
#pragma unroll
    for (int ct = 0; ct < 4; ++ct) {
      v8f c = {};
#pragma unroll
      for (int k = 0; k < 2; ++k) {            // K = DH = 8 in steps of 4
        const int kk = h * DH + k * 4 + lh * 2;
        v2f a; a.x = sQ[row * XP + kk]; a.y = sQ[row * XP + kk + 1];
        const int n = ct * 16 + l16;           // key index
        v2f bbf; bbf.x = sK[n * XP + kk]; bbf.y = sK[n * XP + kk + 1];
        c = wmma4(a, bbf, c);
      }
#pragma unroll
      for (int r = 0; r < 8; ++r)
        sA[(wave * 16 + r + 8 * lh) * AP + ct * 16 + l16] = c[r] * scale;
    }
    __syncthreads();

    // ---- masked softmax: one thread per query row ----
    if (tid < SEQ) {
      float* rp = &sA[tid * AP];
      float m = -3.402823466e38f;
      for (int j = 0; j < SEQ; ++j) {
        const bool ok = (j <= tid) || (j < aa);  // causal OR prefix
        m = fmaxf(m, ok ? rp[j] : -3.402823466e38f);
      }
      float sum = 0.f;
      for (int j = 0; j < SEQ; ++j) {
        const bool ok = (j <= tid) || (j < aa);
        const float e = ok ? __expf(rp[j] - m) : 0.f;
        rp[j] = e;
        sum += e;
      }
      const float inv = 1.0f / sum;
      for (int j = 0; j < SEQ; ++j) rp[j] *= inv;
    }
    __syncthreads();

    // ---- out_h = attn @ Vh : [64,8] (N lanes 8..15 padded with zero) ----
    {
      v8f c = {};
#pragma unroll
      for (int k = 0; k < 16; ++k) {           // K = 64 in steps of 4
        const int kk = k * 4 + lh * 2;         // key index
        v2f a; a.x = sA[row * AP + kk]; a.y = sA[row * AP + kk + 1];
        v2f bbf;
        if (l16 < DH) {                        // B[k][n] = V[k][h*8+n]
          bbf.x = sV[kk * XP + h * DH + l16];
          bbf.y = sV[(kk + 1) * XP + h * DH + l16];
        } else { bbf.x = 0.f; bbf.y = 0.f; }
        c = wmma4(a, bbf, c);                  // uniform: EXEC all-ones here
      }
      if (l16 < DH) {
#pragma unroll
        for (int r = 0; r < 8; ++r)
          sO[(wave * 16 + r + 8 * lh) * XP + h * DH + l16] = c[r];
      }
    }
    __syncthreads();                            // sA reused next head; sO complete at loop exit
  }

  // ---- output projection: Y = O @ Wp^T + bp, streamed to HBM ----
  float* yb = out + (size_t)b * (SEQ * DIM);
#pragma unroll
  for (int nt = 0; nt < 2; ++nt) {
    v8f c = {};
#pragma unroll
    for (int k = 0; k < 8; ++k) {
      const int kk = k * 4 + lh * 2;
      v2f a; a.x = sO[row * XP + kk]; a.y = sO[row * XP + kk + 1];
      const int n = nt * 16 + l16;
      const v2f bbf = *reinterpret_cast<const v2f*>(Wp + n * DIM + kk);
      c = wmma4(a, bbf, c);
    }
    const float bias = bp[nt * 16 + l16];
#pragma unroll
    for (int r = 0; r < 8; ++r)
      yb[(wave * 16 + r + 8 * lh) * DIM + nt * 16 + l16] = c[r] + bias;
  }
}

extern "C" void kernel_launch(void* const* d_in, const int* in_sizes, int n_in,
                              void* d_out, int out_size, void* d_ws, size_t ws_size,
                              hipStream_t stream) {
  const float* x      = (const float*)d_in[0];
  const int*   attend = (const int*)  d_in[1];
  const float* Wq     = (const float*)d_in[2];
  const float* bq     = (const float*)d_in[3];
  const float* Wk     = (const float*)d_in[4];
  const float* bk     = (const float*)d_in[5];
  const float* Wv     = (const float*)d_in[6];
  const float* bv     = (const float*)d_in[7];
  const float* Wp     = (const float*)d_in[8];
  const float* bp     = (const float*)d_in[9];
  float* out = (float*)d_out;

  dim3 grid(NB), block(128);
  hipLaunchKernelGGL(attn_fused_kernel, grid, block, 0, stream,
                     x, attend, Wq, bq, Wk, bk, Wv, bv, Wp, bp, out);
}